// MultiBoxGCNLoss300_42906723287657
// MI455X (gfx1250) — compile-verified
//
#include <hip/hip_runtime.h>
#include <stdint.h>

#define B_   64
#define P_   8732
#define M_   16
#define C_   81
#define NPB  (B_ * P_)          // 558848
#define ROWS 128                // rows per score tile
#define NB2  (NPB / ROWS)       // 4366, exact
#define THRESH 0.5f
#define EPS_   1e-7f

typedef __attribute__((ext_vector_type(4))) unsigned int v4u;
typedef __attribute__((ext_vector_type(8))) int          v8i;
typedef __attribute__((ext_vector_type(4))) int          v4i;

__device__ __forceinline__ float wave_sum_f(float v) {
#pragma unroll
    for (int o = 16; o > 0; o >>= 1) v += __shfl_xor(v, o, 32);
    return v;
}
__device__ __forceinline__ int wave_sum_i(int v) {
#pragma unroll
    for (int o = 16; o > 0; o >>= 1) v += __shfl_xor(v, o, 32);
    return v;
}

// ---------------------------------------------------------------------------
// Kernel 1: prior->object assignment (one workgroup per batch image)
// ---------------------------------------------------------------------------
__global__ void __launch_bounds__(256)
k_assign(const float* __restrict__ boxes, const int* __restrict__ labels,
         const float* __restrict__ priors_cxcy,
         float* __restrict__ ovl, int* __restrict__ objp,
         int* __restrict__ tcls, int* __restrict__ npos)
{
    __shared__ float sbox[M_][4];
    __shared__ int   slab[M_];
    __shared__ float sobjv[M_][256];
    __shared__ int   sobjp[M_][256];
    __shared__ float s_ovl_obj[M_];
    __shared__ int   s_prior_obj[M_];
    __shared__ int   swred[8];

    const int b = blockIdx.x;
    const int tid = threadIdx.x;

    if (tid < M_ * 4) ((float*)sbox)[tid] = boxes[b * M_ * 4 + tid];
    if (tid < M_)     slab[tid] = labels[b * M_ + tid];
    __syncthreads();

    // per-object best prior (thread-local, p ascending -> first-max wins)
    float bestv[M_]; int bestp[M_];
#pragma unroll
    for (int m = 0; m < M_; ++m) { bestv[m] = -1.0f; bestp[m] = 0; }

    for (int p = tid; p < P_; p += 256) {
        float4 pc = ((const float4*)priors_cxcy)[p];
        float px1 = pc.x - pc.z * 0.5f, py1 = pc.y - pc.w * 0.5f;
        float px2 = pc.x + pc.z * 0.5f, py2 = pc.y + pc.w * 0.5f;
        float area_p = (px2 - px1) * (py2 - py1);
        float bo = -1.0f; int bm = 0;
#pragma unroll
        for (int m = 0; m < M_; ++m) {
            float ax1 = sbox[m][0], ay1 = sbox[m][1];
            float ax2 = sbox[m][2], ay2 = sbox[m][3];
            float ltx = fmaxf(ax1, px1), lty = fmaxf(ay1, py1);
            float rbx = fminf(ax2, px2), rby = fminf(ay2, py2);
            float wx = fmaxf(rbx - ltx, 0.0f), wy = fmaxf(rby - lty, 0.0f);
            float inter = wx * wy;
            float area_a = (ax2 - ax1) * (ay2 - ay1);
            float iou = inter / (area_a + area_p - inter);   // faithful: no EPS
            if (iou > bo) { bo = iou; bm = m; }              // first-max (m asc)
            if (iou > bestv[m]) { bestv[m] = iou; bestp[m] = p; }
        }
        ovl[b * P_ + p]  = bo;
        objp[b * P_ + p] = bm;
    }
#pragma unroll
    for (int m = 0; m < M_; ++m) { sobjv[m][tid] = bestv[m]; sobjp[m][tid] = bestp[m]; }
    __syncthreads();

    // cross-thread argmax per object: (max val, min p) lexicographic
    if (tid < M_) {
        float bv = -1.0f; int bp = 0x7fffffff;
        for (int t = 0; t < 256; ++t) {
            float v = sobjv[tid][t]; int p = sobjp[tid][t];
            if (v > bv || (v == bv && p < bp)) { bv = v; bp = p; }
        }
        s_ovl_obj[tid] = bv; s_prior_obj[tid] = bp;
    }
    __threadfence_block();
    __syncthreads();

    // sequential override scatter (matches .at[idx].set last-write order)
    if (tid == 0) {
        int j = -1;
        for (int m = 0; m < M_; ++m) {
            if (s_ovl_obj[m] > 0.0f) {
                ++j;
                int idx = s_prior_obj[m];
                ovl[b * P_ + idx]  = 1.0f;
                objp[b * P_ + idx] = j;    // faithful quirk: filtered index
            }
        }
    }
    __threadfence_block();
    __syncthreads();

    int cnt = 0;
    for (int p = tid; p < P_; p += 256) {
        float o = ovl[b * P_ + p];
        int lab = slab[objp[b * P_ + p]];
        int tc = (o < THRESH) ? 0 : lab;
        tcls[b * P_ + p] = tc;
        cnt += (tc > 0);
    }
    cnt = wave_sum_i(cnt);
    if ((tid & 31) == 0) swred[tid >> 5] = cnt;
    __syncthreads();
    if (tid == 0) {
        int tot = 0;
#pragma unroll
        for (int w = 0; w < 8; ++w) tot += swred[w];
        npos[b] = tot;
    }
}

// ---------------------------------------------------------------------------
// Kernel 2: cross-entropy + fused degenerate DIoU loc term.
// TDM stages a ROWSxC_ fp32 score tile into LDS (tensor_load_to_lds).
// ---------------------------------------------------------------------------
__global__ void __launch_bounds__(ROWS)
k_conf(const float* __restrict__ scores, const float* __restrict__ locs,
       const int* __restrict__ tcls, float* __restrict__ cneg,
       float* __restrict__ cp_part, float* __restrict__ lp_part)
{
    __shared__ float tile[ROWS * C_];
    __shared__ float sredc[4];
    __shared__ float sredl[4];

    const int tid = threadIdx.x;
    const unsigned long long row0 = (unsigned long long)blockIdx.x * ROWS;

    if (tid < 32) {
        unsigned lds_off = (unsigned)(uintptr_t)(&tile[0]);
        unsigned long long ga =
            (unsigned long long)(uintptr_t)scores + row0 * (C_ * 4ull);
        // D# group 0: count=1 | lds_addr | global_addr[56:0] | type=2
        v4u g0 = { 1u,
                   lds_off,
                   (unsigned)(ga & 0xffffffffull),
                   (unsigned)(((ga >> 32) & 0x1ffffffull) | (2u << 30)) };
        // D# group 1: data_size=4B; tensor = tile = C_ x ROWS; stride0 = C_
        v8i g1 = { (int)(2u << 16),                 // data_size code 2 (4B)
                   (int)(((unsigned)C_) << 16),     // tensor_dim0 lo -> [63:48]
                   (int)(((unsigned)ROWS) << 16),   // tensor_dim1 lo -> [95:80]
                   (int)(((unsigned)C_) << 16),     // tile_dim0 -> [127:112]
                   (int)ROWS,                       // tile_dim1 (tile_dim2=0)
                   (int)C_,                         // tensor_dim0_stride lo
                   0, 0 };
        v4i gz4 = { 0, 0, 0, 0 };
        v8i gz8 = { 0, 0, 0, 0, 0, 0, 0, 0 };
        __builtin_amdgcn_tensor_load_to_lds(g0, g1, gz4, gz4, gz8, 0);
        __builtin_amdgcn_s_wait_tensorcnt(0);
    }
    __syncthreads();

    const unsigned long long row = row0 + (unsigned)tid;
    const float* s = &tile[tid * C_];        // stride 81: bank-conflict free

    float m = s[0];
    for (int c = 1; c < C_; ++c) m = fmaxf(m, s[c]);
    float sum = 0.0f;
    for (int c = 0; c < C_; ++c) sum += __expf(s[c] - m);
    float lse = m + __logf(sum);

    const int cls = tcls[row];
    const bool pos = cls > 0;
    const float conf = lse - s[cls];         // -log_softmax[cls]

    // degenerate DIoU(pred,pred): same fp op order as reference
    float4 L = ((const float4*)locs)[row];
    float dx = L.z - L.x, dy = L.w - L.y;
    float wx = fmaxf(dx, 0.0f), wy = fmaxf(dy, 0.0f);
    float inter = wx * wy;
    float area = dx * dy;
    float uni = (area + area) - inter + EPS_;
    float iou = inter / uni;                 // inter_diag term is exactly 0
    float diou = fminf(fmaxf(iou, -1.0f), 1.0f);
    float perbox = 1.0f - diou;

    cneg[row] = pos ? 0.0f : conf;
    float cp = pos ? conf : 0.0f;
    float lp = pos ? perbox : 0.0f;
    cp = wave_sum_f(cp);
    lp = wave_sum_f(lp);
    if ((tid & 31) == 0) { sredc[tid >> 5] = cp; sredl[tid >> 5] = lp; }
    __syncthreads();
    if (tid == 0) {
        cp_part[blockIdx.x] = sredc[0] + sredc[1] + sredc[2] + sredc[3];
        lp_part[blockIdx.x] = sredl[0] + sredl[1] + sredl[2] + sredl[3];
    }
}

// ---------------------------------------------------------------------------
// Kernel 3: hard-negative mining: sum of top-(3*n_pos) conf_neg per batch via
// LDS-resident MSB-first radix select (values are all >= 0).
// ---------------------------------------------------------------------------
__global__ void __launch_bounds__(256)
k_hard(const float* __restrict__ cneg, const int* __restrict__ npos,
       float* __restrict__ hard_part)
{
    __shared__ float vals[P_];
    __shared__ int   sredi[8];
    __shared__ float sredf[8];

    const int b = blockIdx.x;
    const int tid = threadIdx.x;

    int k = 3 * npos[b];
    if (k > P_) k = P_;
    if (k <= 0) { if (tid == 0) hard_part[b] = 0.0f; return; }

    __builtin_prefetch(&cneg[b * P_], 0, 0);
    for (int p = tid; p < P_; p += 256) vals[p] = cneg[b * P_ + p];
    __syncthreads();

    unsigned pref = 0;
    int remaining = k;
    for (int bit = 30; bit >= 0; --bit) {
        const unsigned test = (pref | (1u << bit)) >> bit;
        int c = 0;
        for (int p = tid; p < P_; p += 256)
            c += ((__float_as_uint(vals[p]) >> bit) == test);
        c = wave_sum_i(c);
        if ((tid & 31) == 0) sredi[tid >> 5] = c;
        __syncthreads();
        int tot = 0;
#pragma unroll
        for (int w = 0; w < 8; ++w) tot += sredi[w];
        if (tot >= remaining) pref |= (1u << bit);
        else                  remaining -= tot;
        __syncthreads();
    }
    const float vk = __uint_as_float(pref);   // exact k-th largest

    float ssum = 0.0f;
    for (int p = tid; p < P_; p += 256) {
        float v = vals[p];
        if (v > vk) ssum += v;
    }
    ssum = wave_sum_f(ssum);
    if ((tid & 31) == 0) sredf[tid >> 5] = ssum;
    __syncthreads();
    if (tid == 0) {
        float tot = 0.0f;
#pragma unroll
        for (int w = 0; w < 8; ++w) tot += sredf[w];
        hard_part[b] = tot + (float)remaining * vk;   // ties at vk
    }
}

// ---------------------------------------------------------------------------
// Kernel 4: deterministic final reduction + scalar combine
// ---------------------------------------------------------------------------
__global__ void __launch_bounds__(256)
k_final(const float* __restrict__ cp_part, const float* __restrict__ lp_part,
        const float* __restrict__ hard_part, const int* __restrict__ npos,
        float* __restrict__ out)
{
    __shared__ float rc[256];
    __shared__ float rl[256];
    const int tid = threadIdx.x;

    float cp = 0.0f, lp = 0.0f;
    for (int i = tid; i < NB2; i += 256) { cp += cp_part[i]; lp += lp_part[i]; }
    rc[tid] = cp; rl[tid] = lp;
    __syncthreads();
    for (int s = 128; s > 0; s >>= 1) {          // fixed-order tree: deterministic
        if (tid < s) { rc[tid] += rc[tid + s]; rl[tid] += rl[tid + s]; }
        __syncthreads();
    }
    if (tid == 0) {
        float hard = 0.0f; int np = 0;
        for (int b = 0; b < B_; ++b) { hard += hard_part[b]; np += npos[b]; }
        float nposf = (float)np;
        float conf_loss = (hard + rc[0]) / nposf;           // faithful: no max
        float loc_loss  = rl[0] / fmaxf(nposf, 1.0f);
        out[0] = conf_loss + loc_loss;                      // ALPHA = 1.0
    }
}

// ---------------------------------------------------------------------------
extern "C" void kernel_launch(void* const* d_in, const int* in_sizes, int n_in,
                              void* d_out, int out_size, void* d_ws, size_t ws_size,
                              hipStream_t stream)
{
    const float* locs   = (const float*)d_in[0];   // [B,P,4]
    const float* scores = (const float*)d_in[1];   // [B,P,C]
    const float* boxes  = (const float*)d_in[2];   // [B,M,4]
    const int*   labels = (const int*)d_in[3];     // [B,M]
    const float* priors = (const float*)d_in[4];   // [P,4]
    float* out = (float*)d_out;

    // workspace carve-up (all arrays fully overwritten every call)
    int*   npos = (int*)d_ws;                      // 64
    int*   tcls = npos + 64;                       // NPB
    float* ovl  = (float*)(tcls + NPB);            // NPB
    int*   objp = (int*)(ovl + NPB);               // NPB
    float* cneg = (float*)(objp + NPB);            // NPB
    float* cp_part   = cneg + NPB;                 // NB2
    float* lp_part   = cp_part + NB2;              // NB2
    float* hard_part = lp_part + NB2;              // 64

    k_assign<<<B_, 256, 0, stream>>>(boxes, labels, priors, ovl, objp, tcls, npos);
    k_conf<<<NB2, ROWS, 0, stream>>>(scores, locs, tcls, cneg, cp_part, lp_part);
    k_hard<<<B_, 256, 0, stream>>>(cneg, npos, hard_part);
    k_final<<<1, 256, 0, stream>>>(cp_part, lp_part, hard_part, npos, out);
}